// MultiBoxLoss_19851338842680
// MI455X (gfx1250) — compile-verified
//
#include <hip/hip_runtime.h>
#include <hip/hip_bf16.h>

#define BNUM 128
#define PNUM 8732
#define MNUM 16
#define CNUM 21
#define TPB  256

typedef __attribute__((ext_vector_type(2))) float v2f;
typedef __attribute__((ext_vector_type(8))) float v8f;

// ---------------------------------------------------------------------------
// CDNA5 async global->LDS copy (ASYNCcnt-tracked). saddr form:
//   global_load_async_to_lds_b128 vdst(lds byte off), vaddr(u32 voffset), s[base]
// ---------------------------------------------------------------------------
__device__ __forceinline__ void async_copy_b128(unsigned lds_off, unsigned gl_off,
                                                unsigned long long sbase) {
    asm volatile("global_load_async_to_lds_b128 %0, %1, %2"
                 :: "v"(lds_off), "v"(gl_off), "s"(sbase) : "memory");
}
__device__ __forceinline__ void wait_async0() {
    asm volatile("s_wait_asynccnt 0" ::: "memory");
}

// ---------------------------------------------------------------------------
// Wave32 sum-reduction through the matrix pipe: D = ones(16x4) x B(4x16) + 0
// => every D[m,n] is a column sum of B; total of all 64 B entries is layout-
// independent. Each lane contributes x in one B slot; 4 xor-shuffles finish
// the 16 column sums. Full f32 precision (16x16x4 f32 WMMA).
// ---------------------------------------------------------------------------
__device__ __forceinline__ float wave_reduce_sum(float x) {
#if __has_builtin(__builtin_amdgcn_wmma_f32_16x16x4_f32)
    v2f a; a[0] = 1.0f; a[1] = 1.0f;
    v2f b; b[0] = x;    b[1] = 0.0f;
    v8f c = {};
    c = __builtin_amdgcn_wmma_f32_16x16x4_f32(false, a, false, b, 0, c, false, false);
    float v = c[0];                 // lane holds colsum[lane % 16]
    v += __shfl_xor(v, 8, 32);
    v += __shfl_xor(v, 4, 32);
    v += __shfl_xor(v, 2, 32);
    v += __shfl_xor(v, 1, 32);
    return v;                       // all lanes: sum over the 32 lanes' x
#else
    for (int off = 16; off >= 1; off >>= 1) x += __shfl_xor(x, off, 32);
    return x;
#endif
}

// ---------------------------------------------------------------------------
// Kernel 0: zero per-batch accumulators  acc[b*4+{0:cnt,1:loss_l,2:c_pos,3:c_neg}]
// ---------------------------------------------------------------------------
__global__ void k_init_acc(float* __restrict__ acc) {
    int i = blockIdx.x * blockDim.x + threadIdx.x;
    if (i < BNUM * 4) acc[i] = 0.0f;
}

// ---------------------------------------------------------------------------
// Kernel 1: per-batch best prior for each GT truth (argmax over P of IoU).
// Packed u64 key = (f32 bits of IoU << 32) | (~prior_idx) so ds_max picks the
// max IoU, first index on ties (matches jnp.argmax).
// ---------------------------------------------------------------------------
__global__ void k_match_best_prior(const float* __restrict__ priors,
                                   const float* __restrict__ truths,
                                   int* __restrict__ bpi) {
    const int b = blockIdx.x;
    const int tid = threadIdx.x;
    __shared__ float st[MNUM][4];
    __shared__ float sareaA[MNUM];
    __shared__ unsigned long long sbest[MNUM];
    if (tid < MNUM) {
        const float* t = truths + ((size_t)b * MNUM + tid) * 4;
        float x0 = t[0], y0 = t[1], x1 = t[2], y1 = t[3];
        st[tid][0] = x0; st[tid][1] = y0; st[tid][2] = x1; st[tid][3] = y1;
        sareaA[tid] = (x1 - x0) * (y1 - y0);
        sbest[tid] = 0ull;
    }
    __syncthreads();

    float bv[MNUM]; int bp[MNUM];
#pragma unroll
    for (int m = 0; m < MNUM; ++m) { bv[m] = -1.0f; bp[m] = 0; }

    for (int p = tid; p < PNUM; p += TPB) {
        float4 pr = ((const float4*)priors)[p];
        float bx0 = pr.x - pr.z * 0.5f, by0 = pr.y - pr.w * 0.5f;
        float bx1 = pr.x + pr.z * 0.5f, by1 = pr.y + pr.w * 0.5f;
        float areaB = (bx1 - bx0) * (by1 - by0);
#pragma unroll
        for (int m = 0; m < MNUM; ++m) {
            float ix0 = fmaxf(st[m][0], bx0), iy0 = fmaxf(st[m][1], by0);
            float ix1 = fminf(st[m][2], bx1), iy1 = fminf(st[m][3], by1);
            float iw = fmaxf(ix1 - ix0, 0.0f), ih = fmaxf(iy1 - iy0, 0.0f);
            float inter = iw * ih;
            float iou = inter / (sareaA[m] + areaB - inter);
            if (iou > bv[m]) { bv[m] = iou; bp[m] = p; }
        }
    }
#pragma unroll
    for (int m = 0; m < MNUM; ++m) {
        unsigned hi = __float_as_uint(fmaxf(bv[m], 0.0f));   // IoU >= 0, bits monotone
        unsigned long long key =
            ((unsigned long long)hi << 32) | (unsigned)(0xFFFFFFFFu - (unsigned)bp[m]);
        atomicMax(&sbest[m], key);                           // ds_max_rtn_u64
    }
    __syncthreads();
    if (tid < MNUM)
        bpi[b * MNUM + tid] = (int)(0xFFFFFFFFu - (unsigned)(sbest[tid] & 0xFFFFFFFFu));
}

// ---------------------------------------------------------------------------
// Kernel 2: per-anchor match + override + encode + smooth-L1 + CE (LSE) with
// conf rows staged via async global->LDS b128. Writes loss_c_rank and per-batch
// accumulators.  grid = (ceil(P/256), B)
// ---------------------------------------------------------------------------
__global__ void k_anchor_loss(const float* __restrict__ loc_data,
                              const float* __restrict__ conf_data,
                              const float* __restrict__ priors,
                              const float* __restrict__ truths,
                              const int*   __restrict__ labels,
                              const int*   __restrict__ bpi,
                              float* __restrict__ lcr,
                              float* __restrict__ acc) {
    const int tile = blockIdx.x, b = blockIdx.y;
    const int tid = threadIdx.x;
    __shared__ __align__(16) float confs[TPB * CNUM];  // 21504 B, 16B-aligned tile
    __shared__ float st[MNUM][4];
    __shared__ float sareaA[MNUM];
    __shared__ int   slab[MNUM];
    __shared__ int   sbpi[MNUM];

    const int p0 = tile * TPB;
    const int nt = min(TPB, PNUM - p0);

    // Kick off the async conf-tile copy first (base is 16B aligned: 84*4 | 16*21).
    {
        unsigned long long sbase =
            (unsigned long long)(size_t)(conf_data + ((size_t)b * PNUM + p0) * CNUM);
        unsigned lds_base = (unsigned)(size_t)&confs[0];
        int nbytes = nt * CNUM * 4;
        for (int i = tid * 16; i < nbytes; i += TPB * 16)
            async_copy_b128(lds_base + (unsigned)i, (unsigned)i, sbase);
    }
    if (tid < MNUM) {
        const float* t = truths + ((size_t)b * MNUM + tid) * 4;
        float x0 = t[0], y0 = t[1], x1 = t[2], y1 = t[3];
        st[tid][0] = x0; st[tid][1] = y0; st[tid][2] = x1; st[tid][3] = y1;
        sareaA[tid] = (x1 - x0) * (y1 - y0);
        slab[tid] = labels[b * MNUM + tid];
        sbpi[tid] = bpi[b * MNUM + tid];
    }
    __syncthreads();   // truths/labels ready; async copy still in flight

    const int p = p0 + tid;
    const bool valid = (tid < nt);
    float part_l = 0.0f, part_cpos = 0.0f, part_cnt = 0.0f;
    int conf = 0;
    float4 pr = make_float4(0.f, 0.f, 1.f, 1.f);

    if (valid) {
        pr = ((const float4*)priors)[p];
        float bx0 = pr.x - pr.z * 0.5f, by0 = pr.y - pr.w * 0.5f;
        float bx1 = pr.x + pr.z * 0.5f, by1 = pr.y + pr.w * 0.5f;
        float areaB = (bx1 - bx0) * (by1 - by0);
        float bestv = -1.0f; int besti = 0;
#pragma unroll
        for (int m = 0; m < MNUM; ++m) {
            float ix0 = fmaxf(st[m][0], bx0), iy0 = fmaxf(st[m][1], by0);
            float ix1 = fminf(st[m][2], bx1), iy1 = fminf(st[m][3], by1);
            float iw = fmaxf(ix1 - ix0, 0.0f), ih = fmaxf(iy1 - iy0, 0.0f);
            float inter = iw * ih;
            float iou = inter / (sareaA[m] + areaB - inter);
            if (iou > bestv) { bestv = iou; besti = m; }
        }
#pragma unroll
        for (int m = 0; m < MNUM; ++m)               // force-match each GT's best prior
            if (sbpi[m] == p) { bestv = 2.0f; besti = m; }

        conf = (bestv < 0.5f) ? 0 : (slab[besti] + 1);
        if (conf > 0) {
            float mx0 = st[besti][0], my0 = st[besti][1];
            float mx1 = st[besti][2], my1 = st[besti][3];
            float gcx = ((mx0 + mx1) * 0.5f - pr.x) / (0.1f * pr.z);
            float gcy = ((my0 + my1) * 0.5f - pr.y) / (0.1f * pr.w);
            float gw  = __logf((mx1 - mx0) / pr.z) * 5.0f;   // /0.2
            float gh  = __logf((my1 - my0) / pr.w) * 5.0f;
            float4 ld = ((const float4*)loc_data)[(size_t)b * PNUM + p];
            float d0 = ld.x - gcx, d1 = ld.y - gcy, d2 = ld.z - gw, d3 = ld.w - gh;
            float dd[4] = { d0, d1, d2, d3 };
#pragma unroll
            for (int k = 0; k < 4; ++k) {
                float ad = fabsf(dd[k]);
                part_l += (ad < 1.0f) ? 0.5f * dd[k] * dd[k] : ad - 0.5f;
            }
            part_cnt = 1.0f;
        }
    }

    // Make the staged conf tile visible: each wave drains its own ASYNCcnt,
    // then the barrier publishes across waves.
    wait_async0();
    __syncthreads();

    if (valid) {
        const float* row = &confs[tid * CNUM];      // stride 21: bank-conflict free
        float mx = row[0];
#pragma unroll
        for (int c = 1; c < CNUM; ++c) mx = fmaxf(mx, row[c]);
        float s = 0.0f;
#pragma unroll
        for (int c = 0; c < CNUM; ++c) s += __expf(row[c] - mx);
        float lse = mx + __logf(s);
        float lossc = lse - row[conf];
        float lcr_val;
        if (conf > 0) { part_cpos = lossc; lcr_val = 0.0f; }
        else          { lcr_val = lossc; }
        lcr[(size_t)b * PNUM + p] = lcr_val;
    }

    // Full-EXEC point: reduce the three partials through the matrix pipe.
    float rl = wave_reduce_sum(part_l);
    float rc = wave_reduce_sum(part_cpos);
    float rn = wave_reduce_sum(part_cnt);
    if ((tid & 31) == 0) {
        atomicAdd(&acc[b * 4 + 0], rn);
        atomicAdd(&acc[b * 4 + 1], rl);
        atomicAdd(&acc[b * 4 + 2], rc);
    }
}

// ---------------------------------------------------------------------------
// Kernel 3: per-batch hard-negative mining. Whole 8732-value row lives in LDS;
// 4-pass 8-bit radix-select finds the K-th largest; top-K sum = sum(>T) + ties*T.
// ---------------------------------------------------------------------------
__global__ void k_hardneg(const float* __restrict__ lcr, float* __restrict__ acc) {
    const int b = blockIdx.x;
    const int tid = threadIdx.x;
    __shared__ unsigned vals[PNUM];       // 34928 B (fits easily in 320KB-class LDS)
    __shared__ unsigned hist[256];
    __shared__ unsigned s_pref;
    __shared__ int s_need;
    __shared__ float wsum[8], wcnt[8];

    for (int i = tid; i < PNUM; i += TPB)
        vals[i] = __float_as_uint(lcr[(size_t)b * PNUM + i]);   // all >= 0: bits ordered

    int cnt = (int)(acc[b * 4 + 0] + 0.5f);
    int K = min(3 * cnt, PNUM - 1);
    if (tid == 0) { s_pref = 0u; s_need = K; }
    __syncthreads();

    float negsum = 0.0f;
    if (K > 0) {
        for (int sh = 24; sh >= 0; sh -= 8) {
            hist[tid] = 0u;
            __syncthreads();
            unsigned pref = s_pref;
            for (int i = tid; i < PNUM; i += TPB) {
                unsigned v = vals[i];
                bool ok = (sh == 24) || ((v >> (sh + 8)) == pref);
                if (ok) atomicAdd(&hist[(v >> sh) & 255u], 1u);
            }
            __syncthreads();
            if (tid == 0) {
                int need = s_need; unsigned cum = 0; unsigned pf = s_pref;
                for (int d = 255; d >= 0; --d) {
                    unsigned c = hist[d];
                    if (cum + c >= (unsigned)need) {
                        s_pref = (pf << 8) | (unsigned)d;
                        s_need = need - (int)cum;
                        break;
                    }
                    cum += c;
                }
            }
            __syncthreads();
        }
        unsigned T = s_pref;
        float Tf = __uint_as_float(T);
        float sgt = 0.0f, cgt = 0.0f;
        for (int i = tid; i < PNUM; i += TPB) {
            unsigned v = vals[i];
            if (v > T) { sgt += __uint_as_float(v); cgt += 1.0f; }
        }
        float rs = wave_reduce_sum(sgt);
        float rc = wave_reduce_sum(cgt);
        if ((tid & 31) == 0) { wsum[tid >> 5] = rs; wcnt[tid >> 5] = rc; }
        __syncthreads();
        if (tid == 0) {
            float S = 0.0f, Cg = 0.0f;
            for (int w = 0; w < 8; ++w) { S += wsum[w]; Cg += wcnt[w]; }
            negsum = S + ((float)K - Cg) * Tf;     // include ties at threshold
            acc[b * 4 + 3] = negsum;
        }
    } else if (tid == 0) {
        acc[b * 4 + 3] = 0.0f;
    }
}

// ---------------------------------------------------------------------------
// Kernel 4: final reduction over batches -> two scalars.
// ---------------------------------------------------------------------------
__global__ void k_finalize(const float* __restrict__ acc, float* __restrict__ out) {
    const int tid = threadIdx.x;   // 128 threads, one per batch
    float n  = acc[tid * 4 + 0];
    float ll = acc[tid * 4 + 1];
    float lc = acc[tid * 4 + 2] + acc[tid * 4 + 3];
    for (int off = 16; off >= 1; off >>= 1) {
        n  += __shfl_xor(n,  off, 32);
        ll += __shfl_xor(ll, off, 32);
        lc += __shfl_xor(lc, off, 32);
    }
    __shared__ float sn[4], sl[4], sc[4];
    if ((tid & 31) == 0) { sn[tid >> 5] = n; sl[tid >> 5] = ll; sc[tid >> 5] = lc; }
    __syncthreads();
    if (tid == 0) {
        float N = 0.0f, L = 0.0f, C = 0.0f;
        for (int w = 0; w < 4; ++w) { N += sn[w]; L += sl[w]; C += sc[w]; }
        out[0] = L / N;
        out[1] = C / N;
    }
}

// ---------------------------------------------------------------------------
extern "C" void kernel_launch(void* const* d_in, const int* in_sizes, int n_in,
                              void* d_out, int out_size, void* d_ws, size_t ws_size,
                              hipStream_t stream) {
    (void)in_sizes; (void)n_in; (void)out_size; (void)ws_size;
    const float* loc    = (const float*)d_in[0];   // [B,P,4]
    const float* conf   = (const float*)d_in[1];   // [B,P,21]
    const float* priors = (const float*)d_in[2];   // [P,4]
    const float* truths = (const float*)d_in[3];   // [B,16,4]
    const int*   labels = (const int*)d_in[4];     // [B,16]
    float* out = (float*)d_out;

    char* ws = (char*)d_ws;
    int*   bpi = (int*)ws;                                        // B*16 ints
    float* lcr = (float*)(ws + 8192);                             // B*P floats
    float* acc = (float*)(ws + 8192 + (size_t)BNUM * PNUM * 4);   // B*4 floats

    k_init_acc<<<2, 256, 0, stream>>>(acc);
    k_match_best_prior<<<BNUM, TPB, 0, stream>>>(priors, truths, bpi);
    dim3 gB((PNUM + TPB - 1) / TPB, BNUM);
    k_anchor_loss<<<gB, TPB, 0, stream>>>(loc, conf, priors, truths, labels, bpi, lcr, acc);
    k_hardneg<<<BNUM, TPB, 0, stream>>>(lcr, acc);
    k_finalize<<<1, 128, 0, stream>>>(acc, out);
}